// KDHR_19000935318034
// MI455X (gfx1250) — compile-verified
//
#include <hip/hip_runtime.h>
#include <math.h>

typedef float v2f __attribute__((ext_vector_type(2)));
typedef float v8f __attribute__((ext_vector_type(8)));

#define N_USER 805
#define N_ITEM 390
#define N_SH   1195
#define SH_PAD 1216      // N_SH rounded up to multiple of 64
#define EH_PAD 816       // N_USER rounded up to multiple of 16
#define DIM    64
#define BATCH  16384

// ---------------- utility kernels ----------------

__global__ void zero_f32(float* p, int n) {
  int i = blockIdx.x * blockDim.x + threadIdx.x;
  if (i < n) p[i] = 0.0f;
}

// copy SH_emb (1195x64) into padded Xp (1216x64), zero pad rows
__global__ void pad_copy_rows(const float* __restrict__ src, float* __restrict__ dst,
                              int rows, int rows_pad) {
  int i = blockIdx.x * blockDim.x + threadIdx.x;
  if (i >= rows_pad * DIM) return;
  int row = i / DIM;
  dst[i] = (row < rows) ? src[i] : 0.0f;
}

// Densify the 1M-edge graph into a SH_PAD x SH_PAD count matrix + per-dst counts.
// Counts are small integers -> fp32 atomic adds are exact and order-invariant.
__global__ void build_counts(const int* __restrict__ edges, float* __restrict__ Cmat,
                             float* __restrict__ cnt, int E) {
  int e = blockIdx.x * blockDim.x + threadIdx.x;
  if (e >= E) return;
  int src = edges[e];
  int dst = edges[E + e];
  atomicAdd(&Cmat[(long)dst * SH_PAD + src], 1.0f);
  atomicAdd(&cnt[dst], 1.0f);
}

// ---------------- fp32 WMMA GEMM, B always stored NxK ("nt" form) ----------------
// D = A(MxK) @ B^T where B is NxK with row stride ldb.
// Caller guarantees: M % 64 == 0, B rows readable up to 16-aligned padded N,
// lda/ldb even (8B-aligned float2 loads). N = logical store bound.
// storeTrans=1 writes D transposed: Cout[col*ldc + row] (used to produce NxK
// operands for downstream GEMMs). One 16x16 tile per wave, 4 waves/block.
__global__ void wmma_gemm_nt(const float* __restrict__ A, const float* __restrict__ B,
                             const float* __restrict__ bias, float* __restrict__ Cout,
                             int M, int N, int K, int lda, int ldb, int ldc,
                             int storeTrans) {
  const int lane = threadIdx.x & 31;
  const int wave = threadIdx.x >> 5;
  const int half = lane >> 4;
  const int r    = lane & 15;
  const int n0   = blockIdx.x * 16;
  const int m0   = (blockIdx.y * 4 + wave) * 16;
  const int rowA = m0 + r;
  const int colB = n0 + r;

  const float* Ap = A + (long)rowA * lda + 2 * half;   // 8B aligned
  const float* Bp = B + (long)colB * ldb + 2 * half;   // 8B aligned
  v8f acc = {};
  const int Kmain = K & ~15;

  for (int k0 = 0; k0 < Kmain; k0 += 16) {
#pragma unroll
    for (int u = 0; u < 4; ++u) {
      v2f a = *(const v2f*)(Ap + k0 + 4 * u);
      v2f b = *(const v2f*)(Bp + k0 + 4 * u);
      acc = __builtin_amdgcn_wmma_f32_16x16x4_f32(false, a, false, b,
                                                  (short)0, acc, false, false);
    }
  }

  // K tail (K % 16 != 0): clamped indices + value select -> branch-free loads
  for (int k0 = Kmain; k0 < K; k0 += 4) {
    int ka = k0 + 2 * half;
    int j0 = (ka     < K) ? ka     : K - 1;
    int j1 = (ka + 1 < K) ? ka + 1 : K - 1;
    float a0 = A[(long)rowA * lda + j0];
    float a1 = A[(long)rowA * lda + j1];
    float b0 = B[(long)colB * ldb + j0];
    float b1 = B[(long)colB * ldb + j1];
    v2f a, b;
    a.x = (ka     < K) ? a0 : 0.0f;
    a.y = (ka + 1 < K) ? a1 : 0.0f;
    b.x = (ka     < K) ? b0 : 0.0f;
    b.y = (ka + 1 < K) ? b1 : 0.0f;
    acc = __builtin_amdgcn_wmma_f32_16x16x4_f32(false, a, false, b,
                                                (short)0, acc, false, false);
  }

  float bv = (bias != nullptr && colB < N) ? bias[colB] : 0.0f;
  if (storeTrans) {
#pragma unroll
    for (int i = 0; i < 8; ++i) {
      int row = m0 + i + 8 * half;
      if (colB < N) Cout[(long)colB * ldc + row] = acc[i] + bv;
    }
  } else {
#pragma unroll
    for (int i = 0; i < 8; ++i) {
      int row = m0 + i + 8 * half;
      if (colB < N) Cout[(long)row * ldc + colB] = acc[i] + bv;
    }
  }
}

// ---------------- elementwise / reductions ----------------

__global__ void tanh_div(const float* __restrict__ s, const float* __restrict__ cnt,
                         float* __restrict__ h, int rows) {
  int i = blockIdx.x * blockDim.x + threadIdx.x;
  if (i >= rows * DIM) return;
  int row = i / DIM;
  h[i] = tanhf(s[i] / fmaxf(cnt[row], 1.0f));
}

// per-row L2 normalize (axis=1); dst element (row, j) at dst[row*rs + j*cs]
__global__ void rownorm_write(const float* __restrict__ src, float* __restrict__ dst,
                              int rs, int cs) {
  __shared__ float red[64];
  int row = blockIdx.x;
  int j = threadIdx.x;
  float v = src[(long)row * DIM + j];
  red[j] = v * v;
  __syncthreads();
  for (int o = 32; o > 0; o >>= 1) {
    if (j < o) red[j] += red[j + o];
    __syncthreads();
  }
  float norm = sqrtf(red[0]);
  dst[(long)row * rs + (long)j * cs] = v / fmaxf(norm, 1e-12f);
}

// per-column L2 normalize (axis=0), add into strided dst
__global__ void colnorm_add(const float* __restrict__ src, float* __restrict__ dst,
                            int rows, int rs, int cs) {
  int j = threadIdx.x;   // 64 threads, one per column
  float s = 0.0f;
  for (int i = 0; i < rows; ++i) { float v = src[(long)i * DIM + j]; s += v * v; }
  float inv = 1.0f / fmaxf(sqrtf(s), 1e-12f);
  for (int i = 0; i < rows; ++i)
    dst[(long)i * rs + (long)j * cs] += src[(long)i * DIM + j] * inv;
}

__global__ void rowsum(const float* __restrict__ src, float* __restrict__ dst, int rows, int cols) {
  int i = blockIdx.x * blockDim.x + threadIdx.x;
  if (i >= rows) return;
  const float* p = src + (long)i * cols;
  float s = 0.0f;
  for (int j = 0; j < cols; ++j) s += p[j];
  dst[i] = s;
}

__global__ void div_rows(float* __restrict__ x, const float* __restrict__ d, int rows) {
  int i = blockIdx.x * blockDim.x + threadIdx.x;
  if (i >= rows * DIM) return;
  x[i] = x[i] / d[i / DIM];
}

// one block per column: mean + population variance over BATCH rows
__global__ void bn_stats(const float* __restrict__ z, float* __restrict__ mu, float* __restrict__ var) {
  __shared__ float s1[256], s2[256];
  int j = blockIdx.x;
  int t = threadIdx.x;
  float a = 0.0f, b = 0.0f;
  for (int i = t; i < BATCH; i += 256) {
    float v = z[(long)i * DIM + j];
    a += v; b += v * v;
  }
  s1[t] = a; s2[t] = b;
  __syncthreads();
  for (int o = 128; o > 0; o >>= 1) {
    if (t < o) { s1[t] += s1[t + o]; s2[t] += s2[t + o]; }
    __syncthreads();
  }
  if (t == 0) {
    float m = s1[0] / (float)BATCH;
    mu[j]  = m;
    var[j] = s2[0] / (float)BATCH - m * m;
  }
}

__global__ void bn_relu(float* __restrict__ z, const float* __restrict__ mu,
                        const float* __restrict__ var, const float* __restrict__ gamma,
                        const float* __restrict__ beta) {
  int i = blockIdx.x * blockDim.x + threadIdx.x;
  if (i >= BATCH * DIM) return;
  int j = i & (DIM - 1);
  float v = (z[i] - mu[j]) * rsqrtf(var[j] + 1e-5f) * gamma[j] + beta[j];
  z[i] = v > 0.0f ? v : 0.0f;
}

// ---------------- driver ----------------

extern "C" void kernel_launch(void* const* d_in, const int* in_sizes, int n_in,
                              void* d_out, int out_size, void* d_ws, size_t ws_size,
                              hipStream_t stream) {
  const int*   edges        = (const int*)d_in[1];      // edge_index_SH (2,E)
  const float* prescription = (const float*)d_in[6];    // (BATCH, N_ITEM)
  const float* SH_emb       = (const float*)d_in[7];    // (N_SH, DIM)
  const float* W1   = (const float*)d_in[8];
  const float* b1   = (const float*)d_in[9];
  const float* W2   = (const float*)d_in[10];
  const float* b2   = (const float*)d_in[11];
  const float* mlpW = (const float*)d_in[12];
  const float* mlpb = (const float*)d_in[13];
  const float* gma  = (const float*)d_in[14];
  const float* bta  = (const float*)d_in[15];
  float* out = (float*)d_out;
  const int E = in_sizes[1] / 2;

  // workspace layout (floats), ~16 MB total
  float* ws    = (float*)d_ws;
  float* Cmat  = ws;                            // SH_PAD*SH_PAD   (zeroed)
  float* cnt   = Cmat + (long)SH_PAD * SH_PAD;  // SH_PAD          (zeroed)
  float* eh    = cnt + SH_PAD;                  // EH_PAD*DIM      (zeroed; pad rows stay 0)
  float* esT   = eh + EH_PAD * DIM;             // DIM*N_ITEM  (es transposed, fully written)
  float* Xp    = esT + DIM * N_ITEM;            // SH_PAD*DIM
  float* t0T   = Xp + SH_PAD * DIM;             // DIM*SH_PAD  (t0 transposed)
  float* t1    = t0T + DIM * SH_PAD;            // SH_PAD*DIM
  float* t2    = t1 + SH_PAD * DIM;             // SH_PAD*DIM
  float* esynd = t2 + SH_PAD * DIM;             // BATCH*DIM
  float* psum  = esynd + (long)BATCH * DIM;     // BATCH
  float* zbuf  = psum + BATCH;                  // BATCH*DIM
  float* mu    = zbuf + (long)BATCH * DIM;      // DIM
  float* var   = mu + DIM;                      // DIM

  const int nzero = SH_PAD * SH_PAD + SH_PAD + EH_PAD * DIM;  // Cmat+cnt+eh contiguous
  zero_f32<<<(nzero + 255) / 256, 256, 0, stream>>>(Cmat, nzero);
  pad_copy_rows<<<(SH_PAD * DIM + 255) / 256, 256, 0, stream>>>(SH_emb, Xp, N_SH, SH_PAD);
  build_counts<<<(E + 255) / 256, 256, 0, stream>>>(edges, Cmat, cnt, E);

  dim3 blk(128);
  auto tiles = [](int M, int N) { return dim3((N + 15) / 16, M / 64); };

  // GCN layer 1: t0T = (Xp @ W1^T + b1)^T ; t1 = Cmat @ t0 ; t2 = tanh(t1 / max(cnt,1))
  wmma_gemm_nt<<<tiles(SH_PAD, DIM), blk, 0, stream>>>(Xp, W1, b1, t0T, SH_PAD, DIM, DIM, DIM, DIM, SH_PAD, 1);
  wmma_gemm_nt<<<tiles(SH_PAD, DIM), blk, 0, stream>>>(Cmat, t0T, nullptr, t1, SH_PAD, DIM, SH_PAD, SH_PAD, SH_PAD, DIM, 0);
  tanh_div<<<(SH_PAD * DIM + 255) / 256, 256, 0, stream>>>(t1, cnt, t2, SH_PAD);
  // GCN layer 2
  wmma_gemm_nt<<<tiles(SH_PAD, DIM), blk, 0, stream>>>(t2, W2, b2, t0T, SH_PAD, DIM, DIM, DIM, DIM, SH_PAD, 1);
  wmma_gemm_nt<<<tiles(SH_PAD, DIM), blk, 0, stream>>>(Cmat, t0T, nullptr, t1, SH_PAD, DIM, SH_PAD, SH_PAD, SH_PAD, DIM, 0);
  tanh_div<<<(SH_PAD * DIM + 255) / 256, 256, 0, stream>>>(t1, cnt, t2, SH_PAD);

  // esT = (rownorm(p_item) + colnorm(c_item))^T ; eh = rownorm(p_user) + colnorm(c_user)
  rownorm_write<<<N_ITEM, 64, 0, stream>>>(SH_emb + (long)N_USER * DIM, esT, 1, N_ITEM);
  rownorm_write<<<N_USER, 64, 0, stream>>>(SH_emb, eh, DIM, 1);
  colnorm_add<<<1, 64, 0, stream>>>(t2 + (long)N_USER * DIM, esT, N_ITEM, 1, N_ITEM);
  colnorm_add<<<1, 64, 0, stream>>>(t2, eh, N_USER, DIM, 1);

  // e_synd = prescription @ es ; q = e_synd / rowsum(prescription)
  rowsum<<<(BATCH + 255) / 256, 256, 0, stream>>>(prescription, psum, BATCH, N_ITEM);
  wmma_gemm_nt<<<tiles(BATCH, DIM), blk, 0, stream>>>(prescription, esT, nullptr, esynd, BATCH, DIM, N_ITEM, N_ITEM, N_ITEM, DIM, 0);
  div_rows<<<(BATCH * DIM + 255) / 256, 256, 0, stream>>>(esynd, psum, BATCH);

  // z = q @ mlpW^T + mlpb ; batchnorm ; relu
  wmma_gemm_nt<<<tiles(BATCH, DIM), blk, 0, stream>>>(esynd, mlpW, mlpb, zbuf, BATCH, DIM, DIM, DIM, DIM, DIM, 0);
  bn_stats<<<DIM, 256, 0, stream>>>(zbuf, mu, var);
  bn_relu<<<(BATCH * DIM + 255) / 256, 256, 0, stream>>>(zbuf, mu, var, gma, bta);

  // out = z @ eh^T  (16384 x 805); eh padded to 816 rows (zeros) for checkless loads
  wmma_gemm_nt<<<tiles(BATCH, N_USER), blk, 0, stream>>>(zbuf, eh, nullptr, out, BATCH, N_USER, DIM, DIM, DIM, N_USER, 0);
}